// ICM_7189775253746
// MI455X (gfx1250) — compile-verified
//
#include <hip/hip_runtime.h>

// ---------------------------------------------------------------------------
// ICM product-quantization decode for MI455X (gfx1250, wave32, WMMA).
//
// Phase 1: Pt[i,j,b,k] = C_pair[i,j,k,b]      (16 MB transpose, L2-resident)
// Phase 2: S[n, m*256+k] = -2*(X@C^T) + Cu    (fp32 WMMA 16x16x4, fused epilogue)
// Phase 3: 3 ICM sweeps x 8 codebooks; one wave per point, argmin over K=256
//          via lane-local min + wave32 shfl_xor butterfly (first-index ties).
//
// Workspace layout: [0, 16MB) = Pt ; [16MB, 16MB+268MB) = S.
// ---------------------------------------------------------------------------

typedef float v2f __attribute__((ext_vector_type(2)));
typedef float v8f __attribute__((ext_vector_type(8)));

#define N_PTS   32768
#define DIM     128
#define M_CB    8
#define K_CB    256
#define MK      (M_CB * K_CB)        // 2048
#define ICM_NIT 3
#define PAIR_ELEMS ((size_t)M_CB * M_CB * K_CB * K_CB)   // 4,194,304

// ---------------------------------------------------------------------------
// Kernel 0: transpose pair table so the gathered dimension (old last index b)
// becomes the row index and k is contiguous:  Pt[((i*8+j)*256 + b)*256 + k].
// Output index is the thread's linear id -> fully coalesced writes; strided
// reads are absorbed by L2 (table is 16 MB, read once).
// ---------------------------------------------------------------------------
__global__ void icm_transpose_pair(const float* __restrict__ Cp,
                                   float* __restrict__ Pt) {
    size_t o   = (size_t)blockIdx.x * 256 + threadIdx.x;   // output linear idx
    int    k   = (int)(o & (K_CB - 1));
    size_t r   = o >> 8;
    int    b   = (int)(r & (K_CB - 1));
    size_t ij  = r >> 8;                                    // i*8 + j
    Pt[o] = Cp[(ij * K_CB + k) * K_CB + b];
}

// ---------------------------------------------------------------------------
// Kernel 1: S = -2 * X @ C^T + C_unary, via V_WMMA_F32_16X16X4_F32.
// Block: 256 thr = 8 waves; covers 16 rows (n) x 128 cols. X tile in LDS.
// A operand layout (ISA 7.12.2, 16x4 f32): lanes 0-15 carry K=0/1 in the two
// VGPRs for M=lane; lanes 16-31 carry K=2/3. B mirrors with N=lane&15.
// D layout: acc[r] holds row M = r + 8*(lane>>4), col N = lane&15.
// ---------------------------------------------------------------------------
__global__ void icm_gemm_unary(const float* __restrict__ X,
                               const float* __restrict__ C,
                               const float* __restrict__ Cu,
                               float* __restrict__ S) {
    __shared__ float lX[16 * 129];                 // padded: bank-conflict free

    const int ntile   = blockIdx.x >> 4;           // 0..2047
    const int colgrp  = blockIdx.x & 15;           // 0..15
    const int n_base  = ntile * 16;

    // Cooperative, coalesced load of the 16x128 X tile.
    #pragma unroll
    for (int t = 0; t < 8; ++t) {
        int idx = threadIdx.x + t * 256;           // 0..2047
        int row = idx >> 7;
        int kc  = idx & 127;
        lX[row * 129 + kc] = X[(size_t)(n_base + row) * DIM + kc];
    }
    __syncthreads();

    const int wave  = threadIdx.x >> 5;
    const int lane  = threadIdx.x & 31;
    const int nlo   = lane & 15;                   // M (for A) / N (for B)
    const int khalf = lane >> 4;                   // 0: K={0,1}, 1: K={2,3}
    const int col_base = (colgrp * 8 + wave) * 16; // 16-wide column tile

    const float* Crow = C + (size_t)(col_base + nlo) * DIM;

    v8f acc = {0.f, 0.f, 0.f, 0.f, 0.f, 0.f, 0.f, 0.f};
    #pragma unroll
    for (int kk = 0; kk < 32; ++kk) {              // K = 128 = 32 * 4
        const int k0 = kk * 4 + khalf * 2;
        v2f a, b;
        a.x = lX[nlo * 129 + k0];
        a.y = lX[nlo * 129 + k0 + 1];
        b.x = Crow[k0];
        b.y = Crow[k0 + 1];
        acc = __builtin_amdgcn_wmma_f32_16x16x4_f32(
            /*neg_a=*/false, a, /*neg_b=*/false, b,
            /*c_mod=*/(short)0, acc, /*reuse_a=*/false, /*reuse_b=*/false);
    }

    // Fused epilogue: S = -2*acc + C_unary  (coalesced 16-float row segments).
    const int ncol = col_base + nlo;
    #pragma unroll
    for (int r = 0; r < 8; ++r) {
        const int m = r + 8 * khalf;
        const size_t off = (size_t)(n_base + m) * MK + ncol;
        S[off] = -2.0f * acc[r] + Cu[off];
    }
}

// ---------------------------------------------------------------------------
// Kernel 2: ICM sweeps. One wave per point n. Lane l owns k in [8l, 8l+8).
// Per step: scores = S[n,i,:] + sum_j Pt[i,j,B[n,j],:]; argmin with
// first-occurrence tie-break (lane mapping is k-ascending; butterfly keeps
// the smaller k on equal value). B kept in registers; final state -> d_out.
// ---------------------------------------------------------------------------
__global__ void icm_sweep(const float* __restrict__ S,
                          const float* __restrict__ Pt,
                          const int*   __restrict__ Bin,
                          int*         __restrict__ Bout) {
    const int wave = threadIdx.x >> 5;
    const int lane = threadIdx.x & 31;
    const int n    = blockIdx.x * 8 + wave;

    int b[M_CB];
    #pragma unroll
    for (int j = 0; j < M_CB; ++j) b[j] = Bin[(size_t)n * M_CB + j];

    const float* Sn = S + (size_t)n * MK;
    const int kbase = lane * 8;

    for (int it = 0; it < ICM_NIT; ++it) {
        #pragma unroll
        for (int i = 0; i < M_CB; ++i) {
            const float* Srow = Sn + i * K_CB + kbase;
            float4 a0 = *(const float4*)(Srow);
            float4 a1 = *(const float4*)(Srow + 4);
            #pragma unroll
            for (int j = 0; j < M_CB; ++j) {
                const float* Prow =
                    Pt + ((size_t)(i * M_CB + j) * K_CB + b[j]) * K_CB + kbase;
                float4 p0 = *(const float4*)(Prow);
                float4 p1 = *(const float4*)(Prow + 4);
                a0.x += p0.x; a0.y += p0.y; a0.z += p0.z; a0.w += p0.w;
                a1.x += p1.x; a1.y += p1.y; a1.z += p1.z; a1.w += p1.w;
            }
            // Lane-local argmin over 8 contiguous k (strict < keeps first k).
            float vals[8] = {a0.x, a0.y, a0.z, a0.w, a1.x, a1.y, a1.z, a1.w};
            float bv = vals[0];
            int   bt = 0;
            #pragma unroll
            for (int t = 1; t < 8; ++t)
                if (vals[t] < bv) { bv = vals[t]; bt = t; }
            int bk = kbase + bt;

            // Wave32 butterfly argmin; ties -> smaller k (first occurrence).
            #pragma unroll
            for (int off = 16; off >= 1; off >>= 1) {
                float ov = __shfl_xor(bv, off, 32);
                int   ok = __shfl_xor(bk, off, 32);
                if (ov < bv || (ov == bv && ok < bk)) { bv = ov; bk = ok; }
            }
            b[i] = bk;   // all lanes agree after full butterfly
        }
    }

    if (lane == 0) {
        #pragma unroll
        for (int j = 0; j < M_CB; ++j) Bout[(size_t)n * M_CB + j] = b[j];
    }
}

// ---------------------------------------------------------------------------
// Host entry. Inputs (setup_inputs order): X, C, C_unary, C_pair, B.
// Output: B [32768, 8] int32. Workspace: 16 MB (Pt) + 268 MB (S).
// ---------------------------------------------------------------------------
extern "C" void kernel_launch(void* const* d_in, const int* in_sizes, int n_in,
                              void* d_out, int out_size, void* d_ws, size_t ws_size,
                              hipStream_t stream) {
    const float* X  = (const float*)d_in[0];
    const float* C  = (const float*)d_in[1];
    const float* Cu = (const float*)d_in[2];
    const float* Cp = (const float*)d_in[3];
    const int*   B  = (const int*)d_in[4];
    int* Bout = (int*)d_out;

    float* Pt = (float*)d_ws;                                  // 16 MB
    float* S  = (float*)((char*)d_ws + PAIR_ELEMS * sizeof(float)); // 268 MB

    // Phase 1: transpose pair table (4,194,304 elems / 256 thr).
    icm_transpose_pair<<<(int)(PAIR_ELEMS / 256), 256, 0, stream>>>(Cp, Pt);

    // Phase 2: S = -2*X@C^T + Cu. 2048 n-tiles x 16 col-groups.
    icm_gemm_unary<<<(N_PTS / 16) * 16, 256, 0, stream>>>(X, C, Cu, S);

    // Phase 3: ICM. 8 points per block (one wave each).
    icm_sweep<<<N_PTS / 8, 256, 0, stream>>>(S, Pt, B, Bout);
}